// real_KW5_52699248721960
// MI455X (gfx1250) — compile-verified
//
#include <hip/hip_runtime.h>
#include <cstdint>

// KW distillation distance: per-row L1 between per-head softmax(T=20) and the
// mimic (mean-logit) softmax, averaged over 5 heads. B=16384 rows, C=1000.
//
// Memory-bound: 327.7 MB read -> ~14us at 23.3 TB/s. No matmul -> no WMMA.
// CDNA5 path used: GLOBAL_LOAD_ASYNC_TO_LDS_B128 (ASYNCcnt DMA into LDS),
// wave32 shuffle reductions, ds_load_b128 LDS reads.

#define BLOCK 256

__device__ __forceinline__ float wred_max(float v) {
#pragma unroll
  for (int o = 16; o > 0; o >>= 1) v = fmaxf(v, __shfl_xor(v, o, 32));
  return v;
}
__device__ __forceinline__ float wred_sum(float v) {
#pragma unroll
  for (int o = 16; o > 0; o >>= 1) v += __shfl_xor(v, o, 32);
  return v;
}

__global__ __launch_bounds__(BLOCK) void kw_softmax_l1_kernel(
    const float* __restrict__ o1, const float* __restrict__ o2,
    const float* __restrict__ o3, const float* __restrict__ o4,
    const float* __restrict__ o5, float* __restrict__ out, int C) {
  constexpr float invT = 1.0f / 20.0f;  // temperature 20
  const int row  = blockIdx.x;
  const int t    = threadIdx.x;
  const int lane = t & 31;
  const int wv   = t >> 5;
  const int nq   = C >> 2;  // 250 float4 per head-row (C==1000)

  __shared__ float buf[5][1024];   // 20 KiB row slab (1000 used, padded)
  __shared__ float redMax[8][6];
  __shared__ float redSum[8][6];
  __shared__ float redL1[8];

  const float* heads[5] = {o1, o2, o3, o4, o5};
  const bool active = (t < nq);

  // ---- Stage row slab into LDS via async DMA (bypasses VGPRs) ----
  if (active) {
    unsigned voff = (unsigned)t * 16u;  // byte offset of this lane's float4
#pragma unroll
    for (int h = 0; h < 5; ++h) {
      const float* gbase = heads[h] + (size_t)row * (size_t)C;
      // low 32 bits of a generic pointer to LDS == wave-relative LDS byte addr
      unsigned ldsaddr = (unsigned)(uintptr_t)(&buf[h][4 * t]);
      asm volatile("global_load_async_to_lds_b128 %0, %1, %2"
                   :: "v"(ldsaddr), "v"(voff), "s"(gbase)
                   : "memory");
    }
  }
  asm volatile("s_wait_asynccnt 0" ::: "memory");
  __syncthreads();

  // ---- Pull this thread's 4 columns x 5 heads into registers ----
  float x[5][4];
  float m[4];
  float mx[6];
#pragma unroll
  for (int i = 0; i < 6; ++i) mx[i] = -3.0e38f;

  if (active) {
#pragma unroll
    for (int h = 0; h < 5; ++h) {
      float4 v = *(const float4*)(&buf[h][4 * t]);  // ds_load_b128
      x[h][0] = v.x; x[h][1] = v.y; x[h][2] = v.z; x[h][3] = v.w;
    }
#pragma unroll
    for (int j = 0; j < 4; ++j)
      m[j] = (x[0][j] + x[1][j] + x[2][j] + x[3][j] + x[4][j]) * 0.2f;
#pragma unroll
    for (int h = 0; h < 5; ++h)
#pragma unroll
      for (int j = 0; j < 4; ++j) mx[h] = fmaxf(mx[h], x[h][j]);
#pragma unroll
    for (int j = 0; j < 4; ++j) mx[5] = fmaxf(mx[5], m[j]);
  }

  // ---- Block-reduce the 6 row maxima ----
#pragma unroll
  for (int i = 0; i < 6; ++i) mx[i] = wred_max(mx[i]);
  if (lane == 0) {
#pragma unroll
    for (int i = 0; i < 6; ++i) redMax[wv][i] = mx[i];
  }
  __syncthreads();
  float M[6];
#pragma unroll
  for (int i = 0; i < 6; ++i) {
    float v = -3.0e38f;
#pragma unroll
    for (int w = 0; w < 8; ++w) v = fmaxf(v, redMax[w][i]);
    M[i] = v;
  }

  // ---- exp((x - max)/T) and the 6 partition sums ----
  float e[5][4], em[4];
  float s[6] = {0.f, 0.f, 0.f, 0.f, 0.f, 0.f};
  if (active) {
#pragma unroll
    for (int h = 0; h < 5; ++h) {
#pragma unroll
      for (int j = 0; j < 4; ++j) {
        e[h][j] = __expf((x[h][j] - M[h]) * invT);
        s[h] += e[h][j];
      }
    }
#pragma unroll
    for (int j = 0; j < 4; ++j) {
      em[j] = __expf((m[j] - M[5]) * invT);
      s[5] += em[j];
    }
  }
#pragma unroll
  for (int i = 0; i < 6; ++i) s[i] = wred_sum(s[i]);
  if (lane == 0) {
#pragma unroll
    for (int i = 0; i < 6; ++i) redSum[wv][i] = s[i];
  }
  __syncthreads();
  float S[6];
#pragma unroll
  for (int i = 0; i < 6; ++i) {
    float v = 0.f;
#pragma unroll
    for (int w = 0; w < 8; ++w) v += redSum[w][i];
    S[i] = v;
  }

  // ---- L1 distance of each head's softmax to the mimic softmax ----
  float acc = 0.f;
  if (active) {
    float rS[6];
#pragma unroll
    for (int i = 0; i < 6; ++i) rS[i] = 1.0f / S[i];
#pragma unroll
    for (int j = 0; j < 4; ++j) {
      float pm = em[j] * rS[5];
#pragma unroll
      for (int h = 0; h < 5; ++h) acc += fabsf(e[h][j] * rS[h] - pm);
    }
  }
  acc = wred_sum(acc);
  if (lane == 0) redL1[wv] = acc;
  __syncthreads();

  if (t == 0) {
    float tot = 0.f;
#pragma unroll
    for (int w = 0; w < 8; ++w) tot += redL1[w];
    out[row] = tot * 0.2f;  // average over the 5 heads
  }
}

extern "C" void kernel_launch(void* const* d_in, const int* in_sizes, int n_in,
                              void* d_out, int out_size, void* d_ws, size_t ws_size,
                              hipStream_t stream) {
  (void)n_in; (void)d_ws; (void)ws_size; (void)out_size;
  const int C = 1000;
  const int rows = in_sizes[0] / C;  // 16384
  kw_softmax_l1_kernel<<<rows, BLOCK, 0, stream>>>(
      (const float*)d_in[0], (const float*)d_in[1], (const float*)d_in[2],
      (const float*)d_in[3], (const float*)d_in[4], (float*)d_out, C);
  // d_in[5] is real_KW, empty (size 0): output is just the KW vector.
}